// Cov_10806137716743
// MI455X (gfx1250) — compile-verified
//
#include <hip/hip_runtime.h>
#include <math.h>

typedef __attribute__((ext_vector_type(2))) float v2f;
typedef __attribute__((ext_vector_type(8))) float v8f;
typedef __attribute__((ext_vector_type(4))) unsigned su4;
typedef __attribute__((ext_vector_type(8))) unsigned su8;

#define NROW 8192
#define DDIM 512
#define BM   128
#define BN   128
#define BK   32
#define LDSTR (BK + 2)       // 34: TDM pad (32 dw + 2 dw) -> 8B-aligned, conflict-free frags
#define PANEL (BM * LDSTR)   // floats per LDS panel
#define NCH   (DDIM / BK)
#define EPS  1e-12f

// ---- Tensor Data Mover: 2D tile (BK x BM rows) of f32, row stride DDIM, LDS pad 2 dw/row
__device__ __forceinline__ void tdm_load(unsigned lds_off, unsigned long long gaddr,
                                         su8 g1)
{
    su4 g0;
    g0[0] = 1u;                                     // count=1, user descriptor
    g0[1] = lds_off;                                // lds_addr (bytes)
    g0[2] = (unsigned)gaddr;                        // global_addr[31:0]
    g0[3] = (unsigned)(gaddr >> 32) | (2u << 30);   // global_addr[56:32] | type=2(image)
    asm volatile("tensor_load_to_lds %0, %1" :: "s"(g0), "s"(g1) : "memory");
}

__device__ __forceinline__ su8 tdm_desc_g1()
{
    su8 g1;
    // data_size=2(4B) | pad_enable | pad_interval=4(32 dw) | pad_amount=1(2 dw)
    g1[0] = (2u << 16) | (1u << 20) | (4u << 22) | (1u << 25);
    g1[1] = 0u;                                 // tensor_dim0 = 1<<30 (lo16 = 0)
    g1[2] = 0x4000u;                            // tensor_dim0 hi16 ; tensor_dim1 lo16=0
    g1[3] = 0x4000u | ((unsigned)BK << 16);     // tensor_dim1 hi16 ; tile_dim0 = 32
    g1[4] = (unsigned)BM;                       // tile_dim1 = 128 ; tile_dim2 = 0
    g1[5] = (unsigned)DDIM;                     // tensor_dim0_stride = 512 elements
    g1[6] = 0u;
    g1[7] = 0u;
    return g1;
}

// ---------------- prep: row sum-of-squares, masked a2/b2, init rowmin + sum ----
__global__ __launch_bounds__(256) void prep_kernel(
    const float* __restrict__ seq, const int* __restrict__ qvs,
    const int* __restrict__ sidx, float* __restrict__ a2,
    float* __restrict__ b2, unsigned* __restrict__ rowmin,
    double* __restrict__ sumAcc)
{
    const int row  = blockIdx.x * 8 + (threadIdx.x >> 5);
    const int lane = threadIdx.x & 31;
    const float* p = seq + (size_t)row * DDIM;
    float s = 0.f;
    #pragma unroll
    for (int k = lane; k < DDIM; k += 32) { float v = p[k]; s += v * v; }
    #pragma unroll
    for (int m = 16; m >= 1; m >>= 1) s += __shfl_xor(s, m, 32);
    if (lane == 0) {
        a2[row] = qvs[row] ? s : 0.f;
        b2[row] = sidx[row] ? s : 0.f;
        rowmin[row] = 0x7F800000u;   // +inf bits
    }
    if (blockIdx.x == 0 && threadIdx.x == 0) *sumAcc = 0.0;
}

// ------- fused symmetric Gram (WMMA f32, TDM-fed) + two-sided distance epilogue -------
__global__ __launch_bounds__(256) void dist_kernel(
    const float* __restrict__ seq, const int* __restrict__ qvs,
    const int* __restrict__ sidx, const float* __restrict__ a2g,
    const float* __restrict__ b2g, unsigned* __restrict__ rowmin,
    double* __restrict__ sumAcc)
{
    const int bi = blockIdx.x, bj = blockIdx.y;
    if (bi > bj) return;                       // symmetric: upper triangle only
    const bool diag = (bi == bj);

    __shared__ float As[2 * PANEL];
    __shared__ float Bs[2 * PANEL];
    __shared__ float a2r[BM], b2r[BM], mqr[BM], msr[BM];
    __shared__ float a2c[BN], b2c[BN], mqc[BN], msc[BN];

    const int tid     = threadIdx.x;
    const int rowBase = bi * BM;
    const int colBase = bj * BN;

    for (int i = tid; i < BM; i += 256) {
        const int gr = rowBase + i, gc = colBase + i;
        a2r[i] = a2g[gr];  b2r[i] = b2g[gr];
        mqr[i] = qvs[gr] ? 1.f : 0.f;
        msr[i] = sidx[gr] ? 1.f : 0.f;
        a2c[i] = a2g[gc];  b2c[i] = b2g[gc];
        mqc[i] = qvs[gc] ? 1.f : 0.f;
        msc[i] = sidx[gc] ? 1.f : 0.f;
    }

    const int wave  = tid >> 5;
    const int lane  = tid & 31;
    const int wm    = (wave & 3) * 32;   // 4 waves along M
    const int wn    = (wave >> 2) * 64;  // 2 waves along N
    const int lrow  = lane & 15;
    const int khalf = (lane >> 4) * 2;

    v8f acc[2][4];
    #pragma unroll
    for (int tm = 0; tm < 2; ++tm)
        #pragma unroll
        for (int tn = 0; tn < 4; ++tn)
            acc[tm][tn] = (v8f)0.f;

    const su8 g1 = tdm_desc_g1();
    const unsigned long long seqA =
        (unsigned long long)(uintptr_t)seq + (unsigned long long)rowBase * DDIM * 4ull;
    const unsigned long long seqB =
        (unsigned long long)(uintptr_t)seq + (unsigned long long)colBase * DDIM * 4ull;
    const unsigned ldsA = (unsigned)(uintptr_t)As;
    const unsigned ldsB = (unsigned)(uintptr_t)Bs;

    if (wave == 0) {                       // preload chunk 0 (TDM ignores EXEC; 1 issue/wave)
        tdm_load(ldsA, seqA, g1);
        if (!diag) tdm_load(ldsB, seqB, g1);
    }

    for (int c = 0; c < NCH; ++c) {
        const int cur = c & 1;
        if (wave == 0) {
            if (c + 1 < NCH) {             // prefetch next chunk into the other buffer
                const unsigned long long off = (unsigned long long)(c + 1) * BK * 4ull;
                const unsigned lo = (unsigned)((c + 1) & 1) * PANEL * 4u;
                tdm_load(ldsA + lo, seqA + off, g1);
                if (!diag) {
                    tdm_load(ldsB + lo, seqB + off, g1);
                    __builtin_amdgcn_s_wait_tensorcnt(2);   // current chunk's 2 loads done
                } else {
                    __builtin_amdgcn_s_wait_tensorcnt(1);
                }
            } else {
                __builtin_amdgcn_s_wait_tensorcnt(0);
            }
        }
        __syncthreads();

        const float* Ap = As + cur * PANEL;
        const float* Bp = (diag ? As : Bs) + cur * PANEL;

        #pragma unroll
        for (int ks = 0; ks < BK; ks += 4) {
            v2f aF[2], bF[4];
            #pragma unroll
            for (int tm = 0; tm < 2; ++tm)
                aF[tm] = *(const v2f*)&Ap[(wm + tm * 16 + lrow) * LDSTR + ks + khalf];
            #pragma unroll
            for (int tn = 0; tn < 4; ++tn)
                bF[tn] = *(const v2f*)&Bp[(wn + tn * 16 + lrow) * LDSTR + ks + khalf];
            #pragma unroll
            for (int tm = 0; tm < 2; ++tm)
                #pragma unroll
                for (int tn = 0; tn < 4; ++tn)
                    acc[tm][tn] = __builtin_amdgcn_wmma_f32_16x16x4_f32(
                        false, aF[tm], false, bF[tn],
                        (short)0, acc[tm][tn], false, false);
        }
        __syncthreads();
    }

    // ---- epilogue: masks applied here; both orientations for off-diagonal blocks ----
    const float INF = __uint_as_float(0x7F800000u);
    const int lcol  = lane & 15;
    const int half8 = (lane >> 4) * 8;
    float localSum  = 0.f;

    // orientation 1: i in row-block, j in col-block
    #pragma unroll
    for (int tm = 0; tm < 2; ++tm) {
        #pragma unroll
        for (int v = 0; v < 8; ++v) {
            const int ri   = wm + tm * 16 + half8 + v;
            const float aa = a2r[ri];
            const float mi = 2.0f * mqr[ri];
            float cand = INF;
            #pragma unroll
            for (int tn = 0; tn < 4; ++tn) {
                const int cj = wn + tn * 16 + lcol;
                float d2   = aa + b2c[cj] - mi * msc[cj] * acc[tm][tn][v];
                float dist = sqrtf(fmaxf(d2, EPS));
                localSum  += dist;
                cand = fminf(cand, msc[cj] > 0.f ? dist : INF);
            }
            #pragma unroll
            for (int m = 8; m >= 1; m >>= 1)
                cand = fminf(cand, __shfl_xor(cand, m, 16));
            if (lcol == 0)
                atomicMin(&rowmin[rowBase + ri], __float_as_uint(cand));
        }
    }

    // orientation 2 (transposed tile): i in col-block, j in row-block
    if (!diag) {
        #pragma unroll
        for (int tn = 0; tn < 4; ++tn) {
            const int cj   = wn + tn * 16 + lcol;
            const float ac = a2c[cj];
            const float mc = 2.0f * mqc[cj];
            float candT = INF;
            #pragma unroll
            for (int tm = 0; tm < 2; ++tm) {
                #pragma unroll
                for (int v = 0; v < 8; ++v) {
                    const int ri = wm + tm * 16 + half8 + v;
                    float d2   = ac + b2r[ri] - mc * msr[ri] * acc[tm][tn][v];
                    float dist = sqrtf(fmaxf(d2, EPS));
                    localSum  += dist;
                    candT = fminf(candT, msr[ri] > 0.f ? dist : INF);
                }
            }
            candT = fminf(candT, __shfl_xor(candT, 16, 32));  // merge the two M-halves
            if (lane < 16)
                atomicMin(&rowmin[colBase + cj], __float_as_uint(candT));
        }
    }

    #pragma unroll
    for (int m = 16; m >= 1; m >>= 1) localSum += __shfl_xor(localSum, m, 32);
    if (lane == 0) atomicAdd(sumAcc, (double)localSum);
}

// ---------------- finalize: clamp, 1 - mn/norm, affine --------------------------
__global__ __launch_bounds__(256) void final_kernel(
    const unsigned* __restrict__ rowmin, const double* __restrict__ sumAcc,
    const float* __restrict__ w, const float* __restrict__ bias,
    float* __restrict__ out)
{
    int i = blockIdx.x * 256 + threadIdx.x;
    if (i >= NROW) return;
    float norm = (float)(*sumAcc / ((double)NROW * (double)NROW));
    float mn   = fminf(__uint_as_float(rowmin[i]), norm);
    out[i] = (1.f - mn / norm) * w[0] + bias[0];
}

extern "C" void kernel_launch(void* const* d_in, const int* in_sizes, int n_in,
                              void* d_out, int out_size, void* d_ws, size_t ws_size,
                              hipStream_t stream)
{
    const float* seq  = (const float*)d_in[0];
    const float* w    = (const float*)d_in[1];
    const float* bias = (const float*)d_in[2];
    const int*   qvs  = (const int*)d_in[3];
    const int*   sidx = (const int*)d_in[4];
    float* out = (float*)d_out;

    double*   sumAcc = (double*)d_ws;                // 8B (16B slot)
    float*    a2     = (float*)((char*)d_ws + 16);   // N floats
    float*    b2     = a2 + NROW;                    // N floats
    unsigned* rowmin = (unsigned*)(b2 + NROW);       // N uints

    prep_kernel<<<NROW / 8, 256, 0, stream>>>(seq, qvs, sidx, a2, b2, rowmin, sumAcc);
    dist_kernel<<<dim3(NROW / BM, NROW / BN), 256, 0, stream>>>(
        seq, qvs, sidx, a2, b2, rowmin, sumAcc);
    final_kernel<<<(NROW + 255) / 256, 256, 0, stream>>>(rowmin, sumAcc, w, bias, out);
}